// PostProcessHOI_7206955123049
// MI455X (gfx1250) — compile-verified
//
#include <hip/hip_runtime.h>
#include <hip/hip_bf16.h>
#include <float.h>
#include <stdint.h>

#define NB   64
#define NQ   900
#define NCP1 81
#define NCC  80
#define NV   117
#define QP   1024   // padded to power of two for bitonic sort

__device__ __forceinline__ float sigmoidf_(float x) {
    return 1.0f / (1.0f + __expf(-x));
}

// ---------------------------------------------------------------------------
// Kernel A: fused sigmoid/argmax/mask/score + box transform.
// One wave32 per (b,q) query, 8 waves per 256-thread block.
// All global loads fully unrolled/coalesced so they issue back-to-back
// (no divergent load-wait loops).
// ---------------------------------------------------------------------------
__global__ __launch_bounds__(256)
void hoi_fused_kernel(const float* __restrict__ obj_logits,
                      const float* __restrict__ verb_logits,
                      const float* __restrict__ sub_boxes,
                      const float* __restrict__ obj_boxes,
                      const int*   __restrict__ target_sizes,
                      const float* __restrict__ correct_mat,
                      float* __restrict__ hoi,
                      float* __restrict__ labels_f,
                      float* __restrict__ sb_out,
                      float* __restrict__ ob_out,
                      float* __restrict__ ws_scores)
{
    const int lane = threadIdx.x & 31;
    const int warp = threadIdx.x >> 5;
    const int g    = blockIdx.x * 8 + warp;          // global query id
    if (g >= NB * NQ) return;
    const int b = g / NQ;

    // ---- argmax / max over 81 object logits: 3 coalesced loads, no loop ----
    const float* ol = obj_logits + (size_t)g * NCP1;
    const float l0 = ol[lane];
    const float l1 = ol[lane + 32];                               // 32..63 < 81
    const int   c2 = (lane < NCP1 - 64) ? (lane + 64) : (NCP1-1); // clamp addr
    float       l2 = ol[c2];
    if (lane >= NCP1 - 64) l2 = -FLT_MAX;                         // invalidate dup

    float mval = l0; int midx = lane;
    if (l1 > mval) { mval = l1; midx = lane + 32; }
    if (l2 > mval) { mval = l2; midx = lane + 64; }
    #pragma unroll
    for (int off = 16; off; off >>= 1) {
        float ov = __shfl_xor(mval, off, 32);
        int   oi = __shfl_xor(midx, off, 32);
        if (ov > mval || (ov == mval && oi < midx)) { mval = ov; midx = oi; }
    }
    const float obj_score = sigmoidf_(mval);   // sigmoid monotone -> max(sigmoid)=sigmoid(max)
    const int   label     = midx;              // softmax argmax == logit argmax

    // ---- hoi_scores row (117 verbs): 4 coalesced loads, predicated tail ----
    const float* vl = verb_logits + (size_t)g * NV;
    float* hrow     = hoi + (size_t)g * NV;
    const bool tail = (lane < NV - 96);                           // lane < 21

    const float w0 = vl[lane];
    const float w1 = vl[lane + 32];
    const float w2 = vl[lane + 64];                               // 64..95 < 117
    const int   v3 = tail ? (lane + 96) : (NV - 1);
    const float w3 = vl[v3];

    const bool  has_obj = (label < NCC);
    const float m0 = has_obj ? correct_mat[(lane)      * NCC + label] : 1.0f;
    const float m1 = has_obj ? correct_mat[(lane + 32) * NCC + label] : 1.0f;
    const float m2 = has_obj ? correct_mat[(lane + 64) * NCC + label] : 1.0f;
    const float m3 = has_obj ? correct_mat[(v3)        * NCC + label] : 1.0f;

    const float s0 = sigmoidf_(w0) * obj_score * m0;
    const float s1 = sigmoidf_(w1) * obj_score * m1;
    const float s2 = sigmoidf_(w2) * obj_score * m2;
    const float s3 = sigmoidf_(w3) * obj_score * m3;

    hrow[lane]      = s0;
    hrow[lane + 32] = s1;
    hrow[lane + 64] = s2;
    if (tail) hrow[lane + 96] = s3;

    float mx = fmaxf(fmaxf(s0, s1), s2);
    if (tail) mx = fmaxf(mx, s3);
    #pragma unroll
    for (int off = 16; off; off >>= 1) mx = fmaxf(mx, __shfl_xor(mx, off, 32));

    // ---- lane 0: label, max score, scaled xyxy boxes ----
    if (lane == 0) {
        labels_f[g]  = (float)label;
        ws_scores[g] = mx;
        const float h = (float)target_sizes[b * 2 + 0];
        const float w = (float)target_sizes[b * 2 + 1];
        float4 sbb = ((const float4*)sub_boxes)[g];
        float4 obb = ((const float4*)obj_boxes)[g];
        float4 s4, o4;
        s4.x = (sbb.x - 0.5f * sbb.z) * w;  s4.y = (sbb.y - 0.5f * sbb.w) * h;
        s4.z = (sbb.x + 0.5f * sbb.z) * w;  s4.w = (sbb.y + 0.5f * sbb.w) * h;
        o4.x = (obb.x - 0.5f * obb.z) * w;  o4.y = (obb.y - 0.5f * obb.w) * h;
        o4.z = (obb.x + 0.5f * obb.z) * w;  o4.w = (obb.y + 0.5f * obb.w) * h;
        ((float4*)sb_out)[g] = s4;
        ((float4*)ob_out)[g] = o4;
    }
}

// ---------------------------------------------------------------------------
// Kernel B: per-batch greedy pairwise NMS.
// 1024 threads per block, one block per batch image.
// Bitonic sort in LDS, then CDNA5 async gather (global->LDS, per-lane
// addresses via GLOBAL_LOAD_ASYNC_TO_LDS_B128) of the sorted box data,
// then the sequential suppression sweep.
// ---------------------------------------------------------------------------
__global__ __launch_bounds__(1024)
void hoi_nms_kernel(const float* __restrict__ ws_scores,
                    const float* __restrict__ labels_f,
                    const float* __restrict__ sb_out,
                    const float* __restrict__ ob_out,
                    float* __restrict__ keep_out)
{
    __shared__ float  key[QP];
    __shared__ int    sidx[QP];
    __shared__ float4 sbox[QP];
    __shared__ float4 obox[QP];
    __shared__ float  sarea[QP];
    __shared__ float  oarea[QP];
    __shared__ int    lab[QP];
    __shared__ int    sup[QP];

    const int tid = threadIdx.x;
    const int b   = blockIdx.x;

    key[tid]  = (tid < NQ) ? ws_scores[b * NQ + tid] : -1.0f;  // pads sort last (scores >= 0)
    sidx[tid] = tid;
    __syncthreads();

    // ---- bitonic sort, descending by score (argsort(-max_scores)) ----
    for (int k = 2; k <= QP; k <<= 1) {
        for (int j = k >> 1; j > 0; j >>= 1) {
            const int ixj = tid ^ j;
            if (ixj > tid) {
                const bool up = ((tid & k) == 0);
                float a = key[tid], c = key[ixj];
                if (up ? (a < c) : (a > c)) {
                    key[tid] = c; key[ixj] = a;
                    int t = sidx[tid]; sidx[tid] = sidx[ixj]; sidx[ixj] = t;
                }
            }
            __syncthreads();
        }
    }

    const int orig  = sidx[tid];
    const int origc = (orig < NQ) ? orig : (NQ - 1);   // clamp pad slots to a valid source

    const int labj = (int)labels_f[b * NQ + origc];
    lab[tid] = labj;
    sup[tid] = 0;

    // ---- CDNA5 async gather: sorted float4 boxes straight into LDS ----
    {
        const uint32_t lds_s = (uint32_t)(uintptr_t)&sbox[tid];
        const uint32_t lds_o = (uint32_t)(uintptr_t)&obox[tid];
        const uint64_t gs = (uint64_t)(uintptr_t)(((const float4*)sb_out) + (b * NQ + origc));
        const uint64_t go = (uint64_t)(uintptr_t)(((const float4*)ob_out) + (b * NQ + origc));
        asm volatile("global_load_async_to_lds_b128 %0, %1, off"
                     :: "v"(lds_s), "v"(gs) : "memory");
        asm volatile("global_load_async_to_lds_b128 %0, %1, off"
                     :: "v"(lds_o), "v"(go) : "memory");
        asm volatile("s_wait_asynccnt 0" ::: "memory");
    }
    __syncthreads();

    const float4 sj = sbox[tid];
    const float4 oj = obox[tid];
    const float saj = (sj.z - sj.x + 1.0f) * (sj.w - sj.y + 1.0f);
    const float oaj = (oj.z - oj.x + 1.0f) * (oj.w - oj.y + 1.0f);
    sarea[tid] = saj;
    oarea[tid] = oaj;
    __syncthreads();

    // ---- greedy sweep: step i only writes sup[>i] and reads sup[i] ----
    const bool active = (tid < NQ);
    int supj = 0;                       // this thread's flag, register copy
    for (int i = 0; i < NQ; ++i) {
        if (active && tid > i && !supj && lab[i] == labj && !sup[i]) {
            const float4 si = sbox[i];
            float xx1 = fmaxf(si.x, sj.x), yy1 = fmaxf(si.y, sj.y);
            float xx2 = fminf(si.z, sj.z), yy2 = fminf(si.w, sj.w);
            float w = fmaxf(0.0f, xx2 - xx1 + 1.0f);
            float h = fmaxf(0.0f, yy2 - yy1 + 1.0f);
            float inter = w * h;
            float iou_s = inter / (sarea[i] + saj - inter);

            const float4 oi = obox[i];
            xx1 = fmaxf(oi.x, oj.x); yy1 = fmaxf(oi.y, oj.y);
            xx2 = fminf(oi.z, oj.z); yy2 = fminf(oi.w, oj.w);
            w = fmaxf(0.0f, xx2 - xx1 + 1.0f);
            h = fmaxf(0.0f, yy2 - yy1 + 1.0f);
            inter = w * h;
            float iou_o = inter / (oarea[i] + oaj - inter);

            if (iou_s * sqrtf(iou_o) > 0.7f) { supj = 1; sup[tid] = 1; }
        }
        __syncthreads();
    }

    if (active && orig < NQ) {
        keep_out[b * NQ + orig] = supj ? 0.0f : 1.0f;
    }
}

// ---------------------------------------------------------------------------
extern "C" void kernel_launch(void* const* d_in, const int* in_sizes, int n_in,
                              void* d_out, int out_size, void* d_ws, size_t ws_size,
                              hipStream_t stream)
{
    const float* obj_logits   = (const float*)d_in[0];
    const float* verb_logits  = (const float*)d_in[1];
    const float* sub_boxes    = (const float*)d_in[2];
    const float* obj_boxes    = (const float*)d_in[3];
    const int*   target_sizes = (const int*)  d_in[4];
    const float* correct_mat  = (const float*)d_in[5];

    float* out      = (float*)d_out;
    float* hoi      = out;                               // B*Q*V
    float* labels_f = hoi      + (size_t)NB * NQ * NV;   // B*Q
    float* sb_out   = labels_f + (size_t)NB * NQ;        // B*Q*4
    float* ob_out   = sb_out   + (size_t)NB * NQ * 4;    // B*Q*4
    float* keep_out = ob_out   + (size_t)NB * NQ * 4;    // B*Q

    float* ws_scores = (float*)d_ws;                     // B*Q floats

    const int nquery = NB * NQ;
    hoi_fused_kernel<<<(nquery + 7) / 8, 256, 0, stream>>>(
        obj_logits, verb_logits, sub_boxes, obj_boxes, target_sizes, correct_mat,
        hoi, labels_f, sb_out, ob_out, ws_scores);

    hoi_nms_kernel<<<NB, 1024, 0, stream>>>(
        ws_scores, labels_f, sb_out, ob_out, keep_out);
}